// GeoNetNormalToDepth_76244259438619
// MI455X (gfx1250) — compile-verified
//
#include <hip/hip_runtime.h>
#include <hip/hip_bf16.h>
#include <math.h>

// GeoNet normal->depth refinement, MI455X (gfx1250, wave32).
// Strategy: one 32x32 output tile per 256-thread block (8 wave32s).
// Stage the 40x40 halo (clamped gather == replicate padding) into LDS with
// CDNA5 async global->LDS loads (ASYNCcnt), normalize in place, then run the
// 9x9 stencil entirely out of LDS (ds_load_b128 per tap, 81x reuse).

#define IMG_H 256
#define IMG_W 256
#define TILE 32
#define RAD 4
#define HS (TILE + 2 * RAD) /* 40 */
#define ALPHA 0.95f
#define DEPTH_EPS 1e-5f

#if defined(__has_builtin)
#if __has_builtin(__builtin_amdgcn_global_load_async_to_lds_b32)
#define USE_ASYNC_LDS 1
#endif
#if __has_builtin(__builtin_amdgcn_s_wait_asynccnt)
#define HAVE_WAIT_ASYNCCNT 1
#endif
#endif

typedef __attribute__((address_space(1))) int GInt; // global int
typedef __attribute__((address_space(3))) int LInt; // LDS int

__global__ __launch_bounds__(256) void geonet_n2d_kernel(
    const float* __restrict__ depth, // [B,1,H,W]
    const float* __restrict__ fovp,  // [B]
    const float* __restrict__ nrm,   // [B,3,H,W]
    float* __restrict__ out)         // [B,1,H,W]
{
    __shared__ float4 tile[HS * HS]; // {nx, ny, nz, z} per halo pixel (25.6 KB)

    const int b   = blockIdx.z;
    const int bx0 = blockIdx.x * TILE;
    const int by0 = blockIdx.y * TILE;
    const int tid = threadIdx.y * 32 + threadIdx.x;

    const float fov   = fovp[b];
    const float f     = 2.0f * tanf(0.5f * fov);
    const float inv_f = f * (1.0f / 256.0f); // 1/fx = f/H (H==W==256)

    const float* dpl = depth + (size_t)b * (IMG_H * IMG_W);
    const float* n0  = nrm + ((size_t)b * 3 + 0) * (IMG_H * IMG_W);
    const float* n1  = nrm + ((size_t)b * 3 + 1) * (IMG_H * IMG_W);
    const float* n2  = nrm + ((size_t)b * 3 + 2) * (IMG_H * IMG_W);

    // ---- Phase 1: async-stage halo tile (clamped gather = edge padding) ----
    for (int idx = tid; idx < HS * HS; idx += 256) {
        const int hy = idx / HS;
        const int hx = idx - hy * HS;
        int gy = by0 - RAD + hy;
        gy = min(max(gy, 0), IMG_H - 1);
        int gx = bx0 - RAD + hx;
        gx = min(max(gx, 0), IMG_W - 1);
        const int off = gy * IMG_W + gx;
#if USE_ASYNC_LDS
        float* lp = (float*)&tile[idx];
        __builtin_amdgcn_global_load_async_to_lds_b32((GInt*)(n0 + off),  (LInt*)(lp + 0), 0, 0);
        __builtin_amdgcn_global_load_async_to_lds_b32((GInt*)(n1 + off),  (LInt*)(lp + 1), 0, 0);
        __builtin_amdgcn_global_load_async_to_lds_b32((GInt*)(n2 + off),  (LInt*)(lp + 2), 0, 0);
        __builtin_amdgcn_global_load_async_to_lds_b32((GInt*)(dpl + off), (LInt*)(lp + 3), 0, 0);
#else
        tile[idx] = make_float4(n0[off], n1[off], n2[off], dpl[off]);
#endif
    }
#if USE_ASYNC_LDS
#if HAVE_WAIT_ASYNCCNT
    __builtin_amdgcn_s_wait_asynccnt(0);
#else
    asm volatile("s_wait_asynccnt 0" ::: "memory");
#endif
#endif
    __syncthreads();

    // ---- Phase 2: in-place transform: n = normalize(1 - 2u), z = max(z, eps) ----
    for (int idx = tid; idx < HS * HS; idx += 256) {
        float4 v = tile[idx];
        const float nx = 1.0f - 2.0f * v.x;
        const float ny = 1.0f - 2.0f * v.y;
        const float nz = 1.0f - 2.0f * v.z;
        const float il = 1.0f / sqrtf(nx * nx + ny * ny + nz * nz);
        tile[idx] = make_float4(nx * il, ny * il, nz * il, fmaxf(v.w, DEPTH_EPS));
    }
    __syncthreads();

    // ---- Phase 3: 9x9 kernel regression, 4 pixels per thread ----
    const int lx = threadIdx.x;
#pragma unroll
    for (int i = 0; i < 4; ++i) {
        const int ly = threadIdx.y + 8 * i;
        const int gy = by0 + ly;
        const int gx = bx0 + lx;

        const float4 c = tile[(ly + RAD) * HS + (lx + RAD)];

        // center ray: ((col - cx)/fx, (row - cy)/fy, 1)
        const float rcx    = ((float)gx - 127.5f) * inv_f;
        const float rcy    = ((float)gy - 127.5f) * inv_f;
        const float denom  = c.x * rcx + c.y * rcy + c.z;
        const float invden = 1.0f / denom;

        // neighbor ray components from CLAMPED coords (edge pad is applied to
        // xyz *after* reprojection in the reference)
        float rx[9], ry[9];
#pragma unroll
        for (int d = 0; d < 9; ++d) {
            const int cgx = min(max(gx - RAD + d, 0), IMG_W - 1);
            const int cgy = min(max(gy - RAD + d, 0), IMG_H - 1);
            rx[d] = ((float)cgx - 127.5f) * inv_f;
            ry[d] = ((float)cgy - 127.5f) * inv_f;
        }

        float sv = 0.0f, sw = 0.0f;
        for (int dy = 0; dy < 9; ++dy) {
            const float4* row = &tile[(ly + dy) * HS + lx];
            const float pyr   = ry[dy];
#pragma unroll
            for (int dx = 0; dx < 9; ++dx) {
                const float4 t = row[dx]; // one ds_load_b128
                float w = c.x * t.x + c.y * t.y + c.z * t.z;
                w = (w < ALPHA) ? 0.0f : w; // NaN w stays NaN (matches jnp.where)
                const float px = rx[dx] * t.w;
                const float py = pyr * t.w;
                const float dv = c.x * px + c.y * py + c.z * t.w;
                // accumulate even when w==0 so 0*inf -> NaN propagates like ref
                sv = fmaf(dv * invden, w, sv);
                sw += w;
            }
        }
        float pred = sv / sw;
        pred = (pred != pred) ? 0.001f : pred;           // NaN -> 0.001
        pred = fminf(fmaxf(pred, 0.0f), 1.0f);           // clip [0,1]
        out[(size_t)b * (IMG_H * IMG_W) + gy * IMG_W + gx] = pred;
    }
}

extern "C" void kernel_launch(void* const* d_in, const int* in_sizes, int n_in,
                              void* d_out, int out_size, void* d_ws, size_t ws_size,
                              hipStream_t stream) {
    const float* depth = (const float*)d_in[0]; // [B,1,256,256]
    const float* fov   = (const float*)d_in[1]; // [B]
    const float* nrm   = (const float*)d_in[2]; // [B,3,256,256]
    float* out         = (float*)d_out;

    const int B = in_sizes[1]; // x_fov has one element per batch
    dim3 grid(IMG_W / TILE, IMG_H / TILE, B);
    dim3 block(32, 8, 1);
    geonet_n2d_kernel<<<grid, block, 0, stream>>>(depth, fov, nrm, out);
}